// LocallyConnected2d_87789131530953
// MI455X (gfx1250) — compile-verified
//
#include <hip/hip_runtime.h>
#include <hip/hip_bf16.h>

typedef __attribute__((ext_vector_type(2))) float v2f;
typedef __attribute__((ext_vector_type(8))) float v8f;
typedef __attribute__((ext_vector_type(4))) unsigned int u32x4;
typedef __attribute__((ext_vector_type(8))) int i32x8;
typedef __attribute__((ext_vector_type(4))) int i32x4;

#define W_IN   4096
#define N_ROWS 4096   // BATCH*HEIGHT
#define OUT_C  2048
#define OUT_W  8192
#define KV     36     // input-row band per 16 output rows (scale ~2.0005), padded to 4
#define KB     40     // input-col band per 64 output cols (scale ~0.5), padded
#define KH     12     // input-col band per 16 output cols
#define CUBIC_A (-0.75f)

__device__ __forceinline__ float cubicw(float t) {
    t = fabsf(t);
    if (t <= 1.0f) return ((CUBIC_A + 2.0f) * t - (CUBIC_A + 3.0f)) * t * t + 1.0f;
    if (t <  2.0f) return (((t - 5.0f) * t + 8.0f) * t - 4.0f) * CUBIC_A;
    return 0.0f;
}

// Issue a 2D FP32 tile load (tile0 cols x tile1 rows, row stride = stride0 elements)
// from global `ga` into LDS byte offset `lds_addr` via the Tensor Data Mover.
// Remaining-extent tensor dims give OOB-zero fill at the clamped borders.
__device__ __forceinline__ void tdm_load_2d_f32(unsigned lds_addr, unsigned long long ga,
                                                unsigned td0, unsigned td1,
                                                unsigned tile0, unsigned tile1,
                                                unsigned stride0) {
    u32x4 g0;
    g0.x = 1u;                                   // count=1, is_restore=0, gather off
    g0.y = lds_addr;                             // LDS byte address of tile start
    g0.z = (unsigned)ga;                         // global_addr[31:0]
    g0.w = (unsigned)(ga >> 32) | (2u << 30);    // global_addr[56:32] | type=2
    i32x8 g1;
    g1[0] = 0x20000;                             // wg_mask=0, data_size=2 (4 bytes)
    g1[1] = (int)((td0 & 0xFFFFu) << 16);        // tensor_dim0[15:0]
    g1[2] = (int)(((td0 >> 16) & 0xFFFFu) | ((td1 & 0xFFFFu) << 16));
    g1[3] = (int)(((td1 >> 16) & 0xFFFFu) | (tile0 << 16));   // tile_dim0
    g1[4] = (int)tile1;                          // tile_dim1 (tile_dim2=0 -> 2D)
    g1[5] = (int)stride0;                        // tensor_dim0_stride[31:0]
    g1[6] = 0;
    g1[7] = 0;
    i32x4 g2 = {};                               // unused (2D tile)
    i32x4 g3 = {};
    i32x8 g4 = {};                               // unused trailing group (6-arg form)
    __builtin_amdgcn_tensor_load_to_lds(g0, g1, g2, g3, g4, 0);
}

// ---------------- Stage 1: grouped 3x5 conv + BatchNorm(row) + ReLU ---------
__global__ __launch_bounds__(256) void conv_bn_relu_kernel(
    const float* __restrict__ x, const float* __restrict__ cw,
    const float* __restrict__ cb, const float* __restrict__ gamma,
    const float* __restrict__ beta, float* __restrict__ y) {
    int r = blockIdx.x;              // 0..4095 global row
    int b = r >> 7;                  // block (group) index
    int h = r & 127;                 // row within block
    int tid = threadIdx.x;

    __shared__ float s_w[15];
    __shared__ float s_sum[256];
    __shared__ float s_sq[256];
    __shared__ float s_stats[2];

    if (tid < 15) s_w[tid] = cw[b * 15 + tid];
    __syncthreads();

    float bias = cb[b];
    int c0 = tid * 16;
    float vals[16];
    float lsum = 0.0f, lsq = 0.0f;
#pragma unroll
    for (int i = 0; i < 16; ++i) {
        int c = c0 + i;
        float acc = bias;
#pragma unroll
        for (int dy = 0; dy < 3; ++dy) {
            int hh = h + dy - 1;
            if (hh < 0 || hh > 127) continue;   // zero pad rows (within block)
            const float* row = x + (size_t)(b * 128 + hh) * W_IN;
#pragma unroll
            for (int dx = 0; dx < 5; ++dx) {
                int cc = c + dx - 2;
                float xv = (cc >= 0 && cc < W_IN) ? row[cc] : 0.0f;  // zero pad cols
                acc = fmaf(xv, s_w[dy * 5 + dx], acc);
            }
        }
        vals[i] = acc;
        lsum += acc;
        lsq  = fmaf(acc, acc, lsq);
    }
    s_sum[tid] = lsum;
    s_sq[tid]  = lsq;
    __syncthreads();
    for (int s = 128; s > 0; s >>= 1) {
        if (tid < s) { s_sum[tid] += s_sum[tid + s]; s_sq[tid] += s_sq[tid + s]; }
        __syncthreads();
    }
    if (tid == 0) {
        float mean = s_sum[0] * (1.0f / W_IN);
        float var  = s_sq[0] * (1.0f / W_IN) - mean * mean;   // biased var
        s_stats[0] = mean;
        s_stats[1] = rsqrtf(var + 1e-5f);
    }
    __syncthreads();
    float mean = s_stats[0], inv = s_stats[1];
    float gm = gamma[r], bt = beta[r];
    float* yo = y + (size_t)r * W_IN;
#pragma unroll
    for (int i = 0; i < 16; ++i) {
        float v = fmaf((vals[i] - mean) * inv, gm, bt);
        yo[c0 + i] = fmaxf(v, 0.0f);
    }
}

// -------- Stage 2: vertical bicubic [4096,4096] -> [2048,4096] via FP32 WMMA
// T_tile[16o x 16c] = W[16 x KV] * Y[KV x 16c]; Y band TDM-staged in LDS,
// banded bicubic weights built in LDS while the DMA is in flight.
__global__ __launch_bounds__(128) void resize_rows_wmma_kernel(
    const float* __restrict__ y, float* __restrict__ t) {
    const float s1 = 4095.0f / 2047.0f;
    int g    = blockIdx.y;           // output row group 0..127 (16 rows each)
    int slab = blockIdx.x;           // column slab 0..63 (64 cols each)
    int tid  = threadIdx.x;

    __shared__ float tile[KV][64];   // input-row band x 64 cols (TDM dest)
    __shared__ float Wlds[16][KV];   // banded bicubic weights

    int base = (int)floorf((float)(16 * g) * s1) - 1;
    if (base < 0) base = 0;
    int col0 = slab * 64;

    if (tid < 32) {                  // wave 0 kicks off the DMA first
        unsigned lds_addr = (unsigned)(uintptr_t)&tile[0][0];
        unsigned long long ga =
            (unsigned long long)(uintptr_t)(y + (size_t)base * W_IN + col0);
        tdm_load_2d_f32(lds_addr, ga,
                        (unsigned)(W_IN - col0), (unsigned)(N_ROWS - base),
                        64u, (unsigned)KV, (unsigned)W_IN);
    }
    if (tid < 16) {                  // weights overlap the DMA (clamped taps fold)
        int m = tid;
        for (int k = 0; k < KV; ++k) Wlds[m][k] = 0.0f;
        float pos = (float)(16 * g + m) * s1;
        float x0  = floorf(pos);
        float fr  = pos - x0;
        int ix0   = (int)x0;
#pragma unroll
        for (int j = 0; j < 4; ++j) {
            int idx = ix0 + j - 1;
            idx = idx < 0 ? 0 : (idx > N_ROWS - 1 ? N_ROWS - 1 : idx);
            Wlds[m][idx - base] += cubicw(fr - (float)(j - 1));
        }
    }
    if (tid < 32) __builtin_amdgcn_s_wait_tensorcnt(0);
    __syncthreads();

    int wave = tid >> 5, lane = tid & 31;
    int n  = lane & 15;              // A: output row m; B/D: column n
    int hi = lane >> 4;              // half-wave: k pair {0,1} vs {2,3}

    v8f acc = {};
#pragma unroll
    for (int kc = 0; kc < KV / 4; ++kc) {
        int kk = kc * 4 + hi * 2;
        v2f a, bfrag;
        a.x = Wlds[n][kk];               // adjacent -> ds_load b64
        a.y = Wlds[n][kk + 1];
        bfrag.x = tile[kk][wave * 16 + n];
        bfrag.y = tile[kk + 1][wave * 16 + n];
        acc = __builtin_amdgcn_wmma_f32_16x16x4_f32(
            false, a, false, bfrag, (short)0, acc, false, false);
    }
    int orow = 16 * g + hi * 8;
#pragma unroll
    for (int v = 0; v < 8; ++v)
        t[(size_t)(orow + v) * W_IN + col0 + wave * 16 + n] = acc[v];
}

// -------- Stage 3: horizontal bicubic [2048,4096] -> [2048,8192] via FP32 WMMA
// OUT_tile[16r x 16o] = T[16r x KH] * W[KH x 16o]; T band TDM-staged in LDS,
// per-wave weight blocks built in LDS while the DMA is in flight.
__global__ __launch_bounds__(128) void resize_cols_wmma_kernel(
    const float* __restrict__ t, float* __restrict__ out) {
    const float s2 = 4095.0f / 8191.0f;
    int cblk = blockIdx.x;           // 64 output cols (4 o-groups, one per wave)
    int rblk = blockIdx.y;           // 16 rows
    int tid  = threadIdx.x;
    int r0   = rblk * 16;

    __shared__ float tile[16][KB];      // 16 rows x input-col band (TDM dest)
    __shared__ float Blds[4][KH][16];   // per-wave banded weights

    int base_blk = (int)floorf((float)(cblk * 64) * s2) - 1;
    if (base_blk < 0) base_blk = 0;

    if (tid < 32) {                  // wave 0 kicks off the DMA first
        unsigned lds_addr = (unsigned)(uintptr_t)&tile[0][0];
        unsigned long long ga =
            (unsigned long long)(uintptr_t)(t + (size_t)r0 * W_IN + base_blk);
        tdm_load_2d_f32(lds_addr, ga,
                        (unsigned)(W_IN - base_blk), (unsigned)(OUT_C - r0),
                        (unsigned)KB, 16u, (unsigned)W_IN);
    }
    if (tid < 64) {                  // weights overlap the DMA
        int wv = tid >> 4;           // o-group (wave) 0..3
        int m  = tid & 15;           // output column within group
        int og = cblk * 4 + wv;
        int base_w = (int)floorf((float)(og * 16) * s2) - 1;
        if (base_w < 0) base_w = 0;
        for (int k = 0; k < KH; ++k) Blds[wv][k][m] = 0.0f;
        float pos = (float)(og * 16 + m) * s2;
        float x0  = floorf(pos);
        float fr  = pos - x0;
        int ix0   = (int)x0;
#pragma unroll
        for (int j = 0; j < 4; ++j) {
            int idx = ix0 + j - 1;
            idx = idx < 0 ? 0 : (idx > W_IN - 1 ? W_IN - 1 : idx);
            Blds[wv][idx - base_w][m] += cubicw(fr - (float)(j - 1));
        }
    }
    if (tid < 32) __builtin_amdgcn_s_wait_tensorcnt(0);
    __syncthreads();

    int wave = tid >> 5, lane = tid & 31;
    int n  = lane & 15;              // A: row m; B/D: output col n
    int hi = lane >> 4;
    int og = cblk * 4 + wave;        // this wave's 16-output-col group

    int base_w = (int)floorf((float)(og * 16) * s2) - 1;
    if (base_w < 0) base_w = 0;
    int off = base_w - base_blk;     // band offset inside the LDS tile (<= 25)

    v8f acc = {};
#pragma unroll
    for (int kc = 0; kc < KH / 4; ++kc) {
        int kk = kc * 4 + hi * 2;
        v2f a, bfrag;
        a.x = tile[n][off + kk];         // A: lane m = lane%16 is the row
        a.y = tile[n][off + kk + 1];
        bfrag.x = Blds[wave][kk][n];
        bfrag.y = Blds[wave][kk + 1][n];
        acc = __builtin_amdgcn_wmma_f32_16x16x4_f32(
            false, a, false, bfrag, (short)0, acc, false, false);
    }
#pragma unroll
    for (int v = 0; v < 8; ++v)
        out[(size_t)(r0 + v + hi * 8) * OUT_W + og * 16 + n] = acc[v];
}

extern "C" void kernel_launch(void* const* d_in, const int* in_sizes, int n_in,
                              void* d_out, int out_size, void* d_ws, size_t ws_size,
                              hipStream_t stream) {
    const float* x     = (const float*)d_in[0];   // [4096, 4096]
    const float* cw    = (const float*)d_in[1];   // [32, 1, 3, 5]
    const float* cb    = (const float*)d_in[2];   // [32]
    const float* gamma = (const float*)d_in[3];   // [32, 128]
    const float* beta  = (const float*)d_in[4];   // [32, 128]
    float* out = (float*)d_out;                   // [1, 2048, 8192]

    float* y = (float*)d_ws;                                              // 64 MiB
    float* t = (float*)((char*)d_ws + (size_t)N_ROWS * W_IN * sizeof(float)); // 32 MiB

    conv_bn_relu_kernel<<<N_ROWS, 256, 0, stream>>>(x, cw, cb, gamma, beta, y);
    resize_rows_wmma_kernel<<<dim3(W_IN / 64, OUT_C / 16), 128, 0, stream>>>(y, t);
    resize_cols_wmma_kernel<<<dim3(OUT_W / 64, OUT_C / 16), 128, 0, stream>>>(t, out);
}